// FFCompatibility_43327630082123
// MI455X (gfx1250) — compile-verified
//
#include <hip/hip_runtime.h>
#include <math.h>

// ---- problem constants (match reference) ----
#define NUM_NODES 2000000
#define NUM_FLOPS 800000
#define NBX 168
#define NBY 480
#define NCK 8
#define NCE 8
#define MAP_ELEMS (NBX * NBY * NCK * NCE)   // 5,160,960 floats (~20.6 MB, L2-resident)
#define NUM_BINS  (NBX * NBY)               // 80,640
#define BINS_PER_BLOCK 128                  // one thread per (x,y) bin
#define TILE_DW  (BINS_PER_BLOCK * NCK * NCE)  // 8192 dwords = 32 KB contiguous tile
#define LDS_DW   (TILE_DW + TILE_DW / 64)      // +1 dword pad per 64 (bank-conflict-free)
#define INV_SQRT2F 0.70710678118654752440f

typedef unsigned int u32x4 __attribute__((ext_vector_type(4)));
typedef int          i32x8 __attribute__((ext_vector_type(8)));
typedef int          i32x4 __attribute__((ext_vector_type(4)));

// ---------------- kernel 0: vectorized zero ----------------
__global__ void ffc_zero_kernel(float4* __restrict__ p, int n4) {
    int i = blockIdx.x * blockDim.x + threadIdx.x;
    if (i < n4) p[i] = make_float4(0.f, 0.f, 0.f, 0.f);
}

// ---------------- kernel 1: Gaussian smear scatter ----------------
__global__ void ffc_scatter_kernel(const float* __restrict__ pos,
                                   const float* __restrict__ nsx,
                                   const float* __restrict__ nsy,
                                   const int*   __restrict__ fidx,
                                   const int*   __restrict__ ctrl,
                                   float* __restrict__ dem_map) {
    int f = blockIdx.x * blockDim.x + threadIdx.x;
    if (f >= NUM_FLOPS) return;

    int fi = fidx[f];
    int ck = ctrl[2 * f + 0];
    int ce = ctrl[2 * f + 1];

    float cx = pos[fi] + 0.5f * nsx[fi];
    float cy = pos[NUM_NODES + fi] + 0.5f * nsy[fi];

    int bx0 = min(max((int)floorf(cx), 0), NBX - 1);
    int by0 = min(max((int)floorf(cy), 0), NBY - 1);

    // 6 shared erf edges per dimension (bins bx0-2 .. bx0+2), stddev = 1
    float ex[6], ey[6];
#pragma unroll
    for (int k = 0; k < 6; ++k) {
        ex[k] = erff(((float)(bx0 - 2 + k) - cx) * INV_SQRT2F);
        ey[k] = erff(((float)(by0 - 2 + k) - cy) * INV_SQRT2F);
    }

    const int cc = ck * NCE + ce;  // each flop hits one (ck,ce) plane
#pragma unroll
    for (int i = 0; i < 5; ++i) {
        int bx = bx0 - 2 + i;
        if (bx < 0 || bx >= NBX) continue;
        float dxi = 0.5f * (ex[i + 1] - ex[i]);
#pragma unroll
        for (int j = 0; j < 5; ++j) {
            int by = by0 - 2 + j;
            if (by < 0 || by >= NBY) continue;
            float d = dxi * 0.5f * (ey[j + 1] - ey[j]);  // INST_AREA == 1
            atomicAdd(&dem_map[(bx * NBY + by) * (NCK * NCE) + cc], d);
        }
    }
}

// ---------------- kernel 2: TDM-staged slice quantization ----------------
// 128 bins/block; the 128*64 floats are one contiguous 32 KB run of the map.
// TDM copies it into LDS with 1-dword padding every 64 dwords so the per-lane
// 64-float reads are bank-conflict-free.
__global__ __launch_bounds__(BINS_PER_BLOCK)
void ffc_quantize_kernel(const float* __restrict__ dem_map,
                         float* __restrict__ scale) {
    __shared__ float lds[LDS_DW];
    const unsigned group = blockIdx.x;  // 0 .. NUM_BINS/BINS_PER_BLOCK-1 (exact tiling)
    const unsigned long long gaddr =
        (unsigned long long)(uintptr_t)(dem_map + (size_t)group * TILE_DW);

#if __has_builtin(__builtin_amdgcn_tensor_load_to_lds) && __has_builtin(__builtin_amdgcn_s_wait_tensorcnt)
    if (threadIdx.x < 32u) {            // one wave issues the tensor DMA
        unsigned ldsoff = (unsigned)(uintptr_t)(void*)lds;  // low 32b of flat ptr = LDS offset
        u32x4 g0;
        i32x8 g1;
        i32x4 g2 = {0, 0, 0, 0};
        i32x4 g3 = {0, 0, 0, 0};
        i32x8 g4 = {0, 0, 0, 0, 0, 0, 0, 0};  // extra group (clang-23 6-arg form), zero-filled
        // D# group 0: count=1 (valid), lds_addr, global_addr[56:0], type=2 ("image")
        g0[0] = 1u;
        g0[1] = ldsoff;
        g0[2] = (unsigned)(gaddr & 0xFFFFFFFFull);
        g0[3] = (unsigned)((gaddr >> 32) & 0x01FFFFFFull) | (2u << 30);
        // D# group 1: data_size=4B (2), pad_enable=1, pad_interval=5 (every 64 DW),
        //             pad_amount=0 (1 DW); 1D tile: tensor_dim0=tile_dim0=8192
        g1[0] = (2 << 16) | (1 << 20) | (5 << 22);
        g1[1] = (int)((TILE_DW & 0xFFFF) << 16);       // tensor_dim0[15:0] @ bits 63:48
        g1[2] = (int)((unsigned)TILE_DW >> 16);        // tensor_dim0[31:16]; tensor_dim1=0
        g1[3] = (int)((TILE_DW & 0xFFFF) << 16);       // tile_dim0 @ bits 127:112
        g1[4] = 0;                                     // tile_dim1=0 (1D), tile_dim2=0
        g1[5] = TILE_DW;                               // tensor_dim0_stride (unused for 1D)
        g1[6] = 0;
        g1[7] = 0;
        __builtin_amdgcn_tensor_load_to_lds(g0, g1, g2, g3, g4, 0);
        __builtin_amdgcn_s_wait_tensorcnt(0);
    }
    __syncthreads();
#else
    // fallback: cooperative staged copy with identical padded layout
    for (int k = threadIdx.x; k < TILE_DW; k += BINS_PER_BLOCK)
        lds[k + (k >> 6)] = dem_map[(size_t)group * TILE_DW + k];
    __syncthreads();
#endif

    // thread t owns bin (group*128 + t); its 64 values start at LDS dword t*65
    const int base = (int)threadIdx.x * 65;
    float tot = 0.f, halves = 0.f;
#pragma unroll
    for (int ck = 0; ck < NCK; ++ck) {
        float q = 0.f;
#pragma unroll
        for (int ce = 0; ce < NCE; ++ce) {
            float v = lds[base + ck * NCE + ce];
            tot += v;
            q += ceilf(v * 0.25f);          // quarters: ceil(dem / (CAP/4))
        }
        halves += ceilf(0.5f * q);           // halves
    }
    float slices = ceilf(0.5f * halves);     // slices
    float s = (tot > 0.f) ? slices * 16.f / fmaxf(tot, 1e-12f) : 0.f;
    scale[group * BINS_PER_BLOCK + threadIdx.x] = s;
}

// ---------------- kernel 3: gather scale back to flops ----------------
__global__ void ffc_gather_kernel(const float* __restrict__ pos,
                                  const float* __restrict__ nsx,
                                  const float* __restrict__ nsy,
                                  const int*   __restrict__ fidx,
                                  const float* __restrict__ scale,
                                  float* __restrict__ out) {
    int f = blockIdx.x * blockDim.x + threadIdx.x;
    if (f >= NUM_FLOPS) return;
    int fi = fidx[f];
    float cx = pos[fi] + 0.5f * nsx[fi];
    float cy = pos[NUM_NODES + fi] + 0.5f * nsy[fi];
    int bx0 = min(max((int)floorf(cx), 0), NBX - 1);
    int by0 = min(max((int)floorf(cy), 0), NBY - 1);
    // out = INST_AREA * scale / CAP = scale / 16
    out[fi] = scale[bx0 * NBY + by0] * 0.0625f;
}

extern "C" void kernel_launch(void* const* d_in, const int* in_sizes, int n_in,
                              void* d_out, int out_size, void* d_ws, size_t ws_size,
                              hipStream_t stream) {
    const float* pos  = (const float*)d_in[0];
    const float* nsx  = (const float*)d_in[1];
    const float* nsy  = (const float*)d_in[2];
    const int*   fidx = (const int*)d_in[3];
    const int*   ctrl = (const int*)d_in[4];
    float* out = (float*)d_out;

    float* dem_map = (float*)d_ws;            // MAP_ELEMS floats
    float* scale   = dem_map + MAP_ELEMS;     // NUM_BINS floats

    // 0) zero workspace (map + scale, contiguous) and output
    {
        int n4 = (MAP_ELEMS + NUM_BINS) / 4;  // divisible by 4
        ffc_zero_kernel<<<(n4 + 255) / 256, 256, 0, stream>>>((float4*)d_ws, n4);
    }
    {
        int n4 = NUM_NODES / 4;               // divisible by 4
        ffc_zero_kernel<<<(n4 + 255) / 256, 256, 0, stream>>>((float4*)d_out, n4);
    }
    // 1) smear + scatter (L2-resident float atomics)
    ffc_scatter_kernel<<<(NUM_FLOPS + 255) / 256, 256, 0, stream>>>(
        pos, nsx, nsy, fidx, ctrl, dem_map);
    // 2) slice quantization with TDM staging (exact tiling: 630 blocks)
    ffc_quantize_kernel<<<NUM_BINS / BINS_PER_BLOCK, BINS_PER_BLOCK, 0, stream>>>(
        dem_map, scale);
    // 3) gather per-flop result
    ffc_gather_kernel<<<(NUM_FLOPS + 255) / 256, 256, 0, stream>>>(
        pos, nsx, nsy, fidx, scale, out);
}